// GeometricFeatureEncoder_86431921865221
// MI455X (gfx1250) — compile-verified
//
#include <hip/hip_runtime.h>
#include <hip/hip_bf16.h>
#include <float.h>

// Problem constants from the reference: B=4, N=8192, C=E=64, K=16
#define BB  4
#define NN  8192
#define EE  64
#define KNN 16

// ---------------- CDNA5 WMMA helpers (fp32 16x16x4) --------------------
typedef float v2f __attribute__((ext_vector_type(2)));
typedef float v8f __attribute__((ext_vector_type(8)));

static __device__ __forceinline__ v8f wmma4(v2f a, v2f b, v8f c) {
  // D(16x16,f32) = A(16x4,f32) * B(4x16,f32) + C
  return __builtin_amdgcn_wmma_f32_16x16x4_f32(false, a, false, b, (short)0, c,
                                               false, false);
}

// B-fragment (4x16): VGPR v, lane l -> B[K = v + 2*(l>>4)][N = n0 + (l&15)]
static __device__ __forceinline__ v2f load_bfrag(const float* __restrict__ W,
                                                 int ld, int k0, int n0,
                                                 int lane) {
  int kk = k0 + ((lane >> 4) << 1);
  int nn = n0 + (lane & 15);
  v2f b;
  b.x = W[kk * ld + nn];
  b.y = W[(kk + 1) * ld + nn];
  return b;
}

// =======================================================================
// Kernel 1: exact KNN (K=16, includes self) + PCA normal per point.
// Candidate points stream through LDS in 16KB tiles; per-thread sorted
// top-16 list in registers (compare-bound: the right structure on CDNA5,
// matrix ops would just add an LDS round trip per candidate tile).
// =======================================================================
__global__ __launch_bounds__(256) void knn_normals_kernel(
    const float* __restrict__ points, int* __restrict__ knn_out,
    float4* __restrict__ nrm_out) {
  const int b = blockIdx.y;
  const int n = blockIdx.x * 256 + threadIdx.x;
  const float* __restrict__ pts = points + (size_t)b * NN * 3;

  __shared__ float4 ctile[1024];  // 16 KB candidate tile

  const float qx = pts[n * 3 + 0];
  const float qy = pts[n * 3 + 1];
  const float qz = pts[n * 3 + 2];

  float kd[KNN];
  int   ki[KNN];
#pragma unroll
  for (int j = 0; j < KNN; ++j) { kd[j] = FLT_MAX; ki[j] = 0; }

  for (int t0 = 0; t0 < NN; t0 += 1024) {
    if (t0 + 1024 < NN)  // gfx1250 global_prefetch_b8 on the next tile
      __builtin_prefetch(&pts[(size_t)(t0 + 1024) * 3], 0, 0);
    __syncthreads();
#pragma unroll
    for (int u = 0; u < 4; ++u) {
      int c = threadIdx.x + u * 256;
      int g = t0 + c;
      ctile[c] = make_float4(pts[g * 3 + 0], pts[g * 3 + 1], pts[g * 3 + 2], 0.f);
    }
    __syncthreads();
    for (int c = 0; c < 1024; ++c) {   // LDS broadcast reads
      float4 p = ctile[c];
      float dx = qx - p.x, dy = qy - p.y, dz = qz - p.z;
      float d2 = fmaf(dx, dx, fmaf(dy, dy, dz * dz));
      if (d2 < kd[KNN - 1]) {          // insert (rare after warm-up)
        kd[KNN - 1] = d2; ki[KNN - 1] = t0 + c;
#pragma unroll
        for (int j = KNN - 1; j > 0; --j) {
          if (kd[j] < kd[j - 1]) {
            float td = kd[j]; kd[j] = kd[j - 1]; kd[j - 1] = td;
            int   ti = ki[j]; ki[j] = ki[j - 1]; ki[j - 1] = ti;
          }
        }
      }
    }
  }

  // ---- covariance of the 16 neighbors (gathers hit L0/L2) ----
  float mx = 0.f, my = 0.f, mz = 0.f;
#pragma unroll
  for (int j = 0; j < KNN; ++j) {
    int idx = ki[j];
    mx += pts[idx * 3 + 0]; my += pts[idx * 3 + 1]; mz += pts[idx * 3 + 2];
  }
  mx *= (1.f / KNN); my *= (1.f / KNN); mz *= (1.f / KNN);

  float xx = 0.f, xy = 0.f, xz = 0.f, yy = 0.f, yz = 0.f, zz = 0.f;
#pragma unroll
  for (int j = 0; j < KNN; ++j) {
    int idx = ki[j];
    float ax = pts[idx * 3 + 0] - mx;
    float ay = pts[idx * 3 + 1] - my;
    float az = pts[idx * 3 + 2] - mz;
    xx += ax * ax; xy += ax * ay; xz += ax * az;
    yy += ay * ay; yz += ay * az; zz += az * az;
  }

  // ---- smallest eigenvector of symmetric 3x3 (trig closed form) ----
  float q  = (xx + yy + zz) * (1.f / 3.f);
  float p1 = xy * xy + xz * xz + yz * yz;
  float a0 = xx - q, a1 = yy - q, a2 = zz - q;
  float p2 = a0 * a0 + a1 * a1 + a2 * a2 + 2.f * p1;
  float p  = sqrtf(p2 * (1.f / 6.f)) + 1e-20f;
  float ip = 1.f / p;
  float bxx = a0 * ip, byy = a1 * ip, bzz = a2 * ip;
  float bxy = xy * ip, bxz = xz * ip, byz = yz * ip;
  float detB = bxx * (byy * bzz - byz * byz)
             - bxy * (bxy * bzz - byz * bxz)
             + bxz * (bxy * byz - byy * bxz);
  float rr  = fminf(fmaxf(detB * 0.5f, -1.f), 1.f);
  float phi = acosf(rr) * (1.f / 3.f);
  float lam = q + 2.f * p * cosf(phi + 2.09439510239f);  // smallest eigval

  float r0x = xx - lam, r0y = xy,       r0z = xz;
  float r1x = xy,       r1y = yy - lam, r1z = yz;
  float r2x = xz,       r2y = yz,       r2z = zz - lam;
  float c0x = r0y * r1z - r0z * r1y, c0y = r0z * r1x - r0x * r1z, c0z = r0x * r1y - r0y * r1x;
  float c1x = r0y * r2z - r0z * r2y, c1y = r0z * r2x - r0x * r2z, c1z = r0x * r2y - r0y * r2x;
  float c2x = r1y * r2z - r1z * r2y, c2y = r1z * r2x - r1x * r2z, c2z = r1x * r2y - r1y * r2x;
  float l0 = c0x * c0x + c0y * c0y + c0z * c0z;
  float l1 = c1x * c1x + c1y * c1y + c1z * c1z;
  float l2 = c2x * c2x + c2y * c2y + c2z * c2z;
  float vx = c0x, vy = c0y, vz = c0z, lbest = l0;
  if (l1 > lbest) { vx = c1x; vy = c1y; vz = c1z; lbest = l1; }
  if (l2 > lbest) { vx = c2x; vy = c2y; vz = c2z; lbest = l2; }
  float inv = rsqrtf(lbest + 1e-30f);
  vx *= inv; vy *= inv; vz *= inv;
  // deterministic sign: largest-|component| positive
  float axx = fabsf(vx), ayy = fabsf(vy), azz = fabsf(vz);
  float msel = (axx > ayy) ? ((axx > azz) ? vx : vz) : ((ayy > azz) ? vy : vz);
  if (msel < 0.f) { vx = -vx; vy = -vy; vz = -vz; }

  const size_t row = (size_t)b * NN + n;
#pragma unroll
  for (int j = 0; j < KNN; ++j) knn_out[row * KNN + j] = ki[j];
  nrm_out[row] = make_float4(vx, vy, vz, 0.f);
}

// =======================================================================
// Kernel 2: curvature = 1 - mean_j dot(n_i, n_knn[j]); builds geo features
// (needs ALL normals -> separate kernel = grid-wide sync)
// =======================================================================
__global__ __launch_bounds__(256) void curvature_kernel(
    const int* __restrict__ knn, const float4* __restrict__ nrm,
    float4* __restrict__ geo) {
  const int r = blockIdx.x * 256 + threadIdx.x;  // [0, B*N)
  const int base = (r / NN) * NN;
  float4 nm = nrm[r];
  float s = 0.f;
#pragma unroll
  for (int j = 0; j < KNN; ++j) {
    int idx = knn[(size_t)r * KNN + j];
    float4 nj = nrm[base + idx];
    s += nm.x * nj.x + nm.y * nj.y + nm.z * nj.z;
  }
  geo[r] = make_float4(nm.x, nm.y, nm.z, 1.f - s * (1.f / KNN));
}

// =======================================================================
// Kernel 3: fused MLP chain in fp32 on V_WMMA_F32_16X16X4_F32.
// One wave per 16-row tile; 226 WMMAs per tile; weights stream from the
// 192MB L2 (57KB total, shared by all WGs); per-wave 8KB LDS staging for
// the C-layout -> A-layout conversion between chained GEMMs.
// =======================================================================
__global__ __launch_bounds__(128) void fused_mlp_wmma_kernel(
    const float* __restrict__ sem, const float* __restrict__ geo,
    const float* __restrict__ Wg1, const float* __restrict__ bg1,
    const float* __restrict__ Wg2, const float* __restrict__ bg2,
    const float* __restrict__ Wgate, const float* __restrict__ bgate,
    const float* __restrict__ Wsem, const float* __restrict__ bsem,
    float* __restrict__ out) {
  __shared__ __align__(16) float stage[4][16 * 128];  // 8KB per wave

  const int wave = threadIdx.x >> 5;
  const int lane = threadIdx.x & 31;
  const int lrow = lane & 15;   // column / A-row selector
  const int lhi  = lane >> 4;   // half selector
  const int tile = blockIdx.x * 4 + wave;
  const int r0   = tile * 16;
  float* st = stage[wave];

  // ---- geo hidden: H1(16x32) = relu(G(16x4) @ Wg1(4x32) + b1) ----
  v2f ag;
  {
    const float* gp = geo + (size_t)(r0 + lrow) * 4 + 2 * lhi;
    ag.x = gp[0]; ag.y = gp[1];
  }
#pragma unroll
  for (int nt = 0; nt < 2; ++nt) {
    v2f bf = load_bfrag(Wg1, 32, 0, nt * 16, lane);
    v8f c = {};
    c = wmma4(ag, bf, c);
    float bb = bg1[nt * 16 + lrow];
#pragma unroll
    for (int r = 0; r < 8; ++r) {
      float v = c[r] + bb;
      st[(r + 8 * lhi) * 32 + nt * 16 + lrow] = v > 0.f ? v : 0.f;
    }
  }
  __syncthreads();

  // ---- geo_embed(16x64) = H1 @ Wg2(32x64) + b2 ----
  v8f ce[4] = {{}, {}, {}, {}};
#pragma unroll
  for (int kc = 0; kc < 8; ++kc) {
    v2f a;
    const float* ap = &st[lrow * 32 + kc * 4 + 2 * lhi];
    a.x = ap[0]; a.y = ap[1];
#pragma unroll
    for (int nt = 0; nt < 4; ++nt) {
      v2f bf = load_bfrag(Wg2, 64, kc * 4, nt * 16, lane);
      ce[nt] = wmma4(a, bf, ce[nt]);
    }
  }
#pragma unroll
  for (int nt = 0; nt < 4; ++nt) {
    float bb = bg2[nt * 16 + lrow];
#pragma unroll
    for (int r = 0; r < 8; ++r) ce[nt][r] += bb;
  }

  // ---- sem_embed(16x64) = S(16x64) @ Wsem(64x64) + bsem ----
  v8f cs[4] = {{}, {}, {}, {}};
  const float* semrow = sem + (size_t)(r0 + lrow) * 64;
#pragma unroll
  for (int kc = 0; kc < 16; ++kc) {
    v2f a;
    a.x = semrow[kc * 4 + 2 * lhi];
    a.y = semrow[kc * 4 + 2 * lhi + 1];
#pragma unroll
    for (int nt = 0; nt < 4; ++nt) {
      v2f bf = load_bfrag(Wsem, 64, kc * 4, nt * 16, lane);
      cs[nt] = wmma4(a, bf, cs[nt]);
    }
  }
#pragma unroll
  for (int nt = 0; nt < 4; ++nt) {
    float bb = bsem[nt * 16 + lrow];
#pragma unroll
    for (int r = 0; r < 8; ++r) cs[nt][r] += bb;
  }

  // ---- stage concat [geo_embed | sem_embed] (16x128) ----
  __syncthreads();
#pragma unroll
  for (int nt = 0; nt < 4; ++nt)
#pragma unroll
    for (int r = 0; r < 8; ++r) {
      st[(r + 8 * lhi) * 128 + nt * 16 + lrow]      = ce[nt][r];
      st[(r + 8 * lhi) * 128 + 64 + nt * 16 + lrow] = cs[nt][r];
    }
  __syncthreads();

  // ---- gate logits (16x64) = concat @ Wgate(128x64) + bgate ----
  v8f cg[4] = {{}, {}, {}, {}};
#pragma unroll
  for (int kc = 0; kc < 32; ++kc) {
    v2f a;
    const float* ap = &st[lrow * 128 + kc * 4 + 2 * lhi];
    a.x = ap[0]; a.y = ap[1];
#pragma unroll
    for (int nt = 0; nt < 4; ++nt) {
      v2f bf = load_bfrag(Wgate, 64, kc * 4, nt * 16, lane);
      cg[nt] = wmma4(a, bf, cg[nt]);
    }
  }

  // ---- sigmoid gate + blend + store ----
#pragma unroll
  for (int nt = 0; nt < 4; ++nt) {
    float bb = bgate[nt * 16 + lrow];
#pragma unroll
    for (int r = 0; r < 8; ++r) {
      float g = 1.f / (1.f + __expf(-(cg[nt][r] + bb)));
      float o = g * cs[nt][r] + (1.f - g) * ce[nt][r];
      out[(size_t)(r0 + r + 8 * lhi) * 64 + nt * 16 + lrow] = o;
    }
  }
}

// =======================================================================
extern "C" void kernel_launch(void* const* d_in, const int* in_sizes, int n_in,
                              void* d_out, int out_size, void* d_ws,
                              size_t ws_size, hipStream_t stream) {
  const float* points = (const float*)d_in[0];
  const float* sem    = (const float*)d_in[1];
  const float* Wg1    = (const float*)d_in[2];
  const float* bg1    = (const float*)d_in[3];
  const float* Wg2    = (const float*)d_in[4];
  const float* bg2    = (const float*)d_in[5];
  const float* Wgate  = (const float*)d_in[6];
  const float* bgate  = (const float*)d_in[7];
  const float* Wsem   = (const float*)d_in[8];
  const float* bsem   = (const float*)d_in[9];
  float* out = (float*)d_out;

  char* ws = (char*)d_ws;
  int*    knn = (int*)ws;                                   // B*N*K ints  (2MB)
  float4* nrm = (float4*)(ws + (size_t)BB * NN * KNN * 4);  // B*N float4  (512KB)
  float4* geo = (float4*)(ws + (size_t)BB * NN * KNN * 4
                             + (size_t)BB * NN * sizeof(float4));

  dim3 g1(NN / 256, BB);
  knn_normals_kernel<<<g1, 256, 0, stream>>>(points, knn, nrm);
  curvature_kernel<<<(BB * NN) / 256, 256, 0, stream>>>(knn, nrm, geo);
  fused_mlp_wmma_kernel<<<(BB * NN / 16) / 4, 128, 0, stream>>>(
      sem, (const float*)geo, Wg1, bg1, Wg2, bg2, Wgate, bgate, Wsem, bsem,
      out);
}